// DCT_26688926778120
// MI455X (gfx1250) — compile-verified
//
#include <hip/hip_runtime.h>
#include <math.h>

typedef __attribute__((ext_vector_type(2))) float v2f;
typedef __attribute__((ext_vector_type(8))) float v8f;

// Input  x: (B=8, C=3, T=32, H=256, W=256) f32
// Output  : (B=8, T=32, P=1024, 8, 8) f32
#define CH_STRIDE  2097152L   // 32*256*256
#define B_STRIDE   6291456L   // 3*CH_STRIDE
#define FRAME_HW   65536L     // 256*256

__global__ __launch_bounds__(256) void dct8x8_wmma_kernel(
    const float* __restrict__ x, float* __restrict__ out)
{
    // ---- 8x8 DCT-II matrix D[k][n] = 2*cos(pi*(2n+1)*k/16), built once per block
    __shared__ float Dtab[64];
    const int tid = threadIdx.x;
    if (tid < 64) {
        const int k = tid >> 3, n = tid & 7;
        Dtab[tid] = 2.0f * cosf(0.19634954084936207f * (float)((2 * n + 1) * k));
    }
    __syncthreads();

    const int wave = tid >> 5;
    const int lane = tid & 31;
    const int tile = blockIdx.x * 8 + wave;   // 65536 tiles total
    const int frame = tile >> 8;              // b*32 + t  (0..255)
    const int t16   = tile & 255;
    const int ty = t16 >> 4, tx = t16 & 15;   // 16x16 tile coords in 256x256 frame

    const int hl    = lane >> 4;              // which half-wave (0/1)
    const int col16 = lane & 15;              // N / M index inside 16x16
    const int i8    = lane & 7;               // DCT frequency row for constants
    const int hl8   = (lane >> 3) & 1;        // block-diagonal mask selector

    // ---- Constant fragments: Wc[kk] serves BOTH step-1 B (=BD^T) and step-2 A (=BD)
    // element = BD[lane&15][K], K = 4*kk + 2*hl + v, masked by block-diagonal.
    v2f Wc[4];
#pragma unroll
    for (int kk = 0; kk < 4; ++kk) {
#pragma unroll
        for (int v = 0; v < 2; ++v) {
            const float d = Dtab[i8 * 8 + 4 * (kk & 1) + 2 * hl + v];
            Wc[kk][v] = ((kk >> 1) == hl8) ? d : 0.0f;
        }
    }

    // ---- Load grayscale 16x16 tile directly in A-fragment layout.
    // A[M][K]: M = lane&15 (tile row), K = 4*kk + 2*hl + v (tile col)
    const int  row  = 16 * ty + col16;
    const long fb   = (long)(frame >> 5) * B_STRIDE + (long)(frame & 31) * FRAME_HW
                    + (long)row * 256 + (long)(16 * tx);
    v2f Ga[4];
#pragma unroll
    for (int kk = 0; kk < 4; ++kk) {
        const long base = fb + 4 * kk + 2 * hl;
        const v2f r = *(const v2f*)(x + base);
        const v2f g = *(const v2f*)(x + base + CH_STRIDE);
        const v2f b = *(const v2f*)(x + base + 2 * CH_STRIDE);
        Ga[kk].x = fmaf(0.114f, b.x, fmaf(0.587f, g.x, 0.2989f * r.x));
        Ga[kk].y = fmaf(0.114f, b.y, fmaf(0.587f, g.y, 0.2989f * r.y));
    }

    // ---- Step 1: P = G @ BD^T   (four K=4 WMMA ops, f32 exact)
    v8f P = {0.f, 0.f, 0.f, 0.f, 0.f, 0.f, 0.f, 0.f};
#pragma unroll
    for (int kk = 0; kk < 4; ++kk) {
        P = __builtin_amdgcn_wmma_f32_16x16x4_f32(
                false, Ga[kk], false, Wc[kk], (short)0, P, false, false);
    }

    // ---- Convert P from C/D layout to B-operand layout.
    // Need P[K][N], K = 4*kk + 2*hl + v; source reg = 4*(kk&1)+2*hl+v,
    // source half = kk>>1  -> half-wave exchange via shfl_xor(16).
    v2f Pb[4];
#pragma unroll
    for (int p = 0; p < 2; ++p) {       // p = kk&1 -> C/D reg base 4*p
#pragma unroll
        for (int v = 0; v < 2; ++v) {
            const float lo   = P[4 * p + v];        // row-half 0 resident value
            const float hi   = P[4 * p + 2 + v];    // row-half 1 resident value
            const float send = hl ? lo : hi;        // what the partner lane needs
            const float own  = hl ? hi : lo;        // what we keep locally
            const float recv = __shfl_xor(send, 16, 32);
            Pb[p][v]     = hl ? recv : own;         // kk = p   (source half 0)
            Pb[p + 2][v] = hl ? own : recv;         // kk = p+2 (source half 1)
        }
    }

    // ---- Step 2: Out = BD @ P
    v8f acc = {0.f, 0.f, 0.f, 0.f, 0.f, 0.f, 0.f, 0.f};
#pragma unroll
    for (int kk = 0; kk < 4; ++kk) {
        acc = __builtin_amdgcn_wmma_f32_16x16x4_f32(
                false, Wc[kk], false, Pb[kk], (short)0, acc, false, false);
    }

    // ---- Store: C/D layout lane holds OutTile[r = v'+8*hl][c = col16].
    // Output block p = bh*32 + bw, within-block (k, l2): bh = 2*ty+hl,
    // bw = 2*tx + (c>>3), k = v', l2 = c&7.
    const long obase = (long)frame * 65536L
                     + (long)(((2 * ty + hl) * 32 + 2 * tx + (col16 >> 3)) * 64 + (col16 & 7));
#pragma unroll
    for (int v = 0; v < 8; ++v) {
        out[obase + v * 8] = acc[v];
    }
}

extern "C" void kernel_launch(void* const* d_in, const int* in_sizes, int n_in,
                              void* d_out, int out_size, void* d_ws, size_t ws_size,
                              hipStream_t stream)
{
    const float* x = (const float*)d_in[0];
    float* out = (float*)d_out;
    // 256 frames * 256 tiles/frame = 65536 tiles; 1 wave (32 threads) per tile,
    // 8 waves per 256-thread block -> 8192 blocks.
    dct8x8_wmma_kernel<<<dim3(8192), dim3(256), 0, stream>>>(x, out);
}